// ChebNetConv_34531537059970
// MI455X (gfx1250) — compile-verified
//
#include <hip/hip_runtime.h>
#include <hip/hip_bf16.h>

// ChebNetConv for MI455X (gfx1250, wave32).
// Pipeline: 3x edge-scatter SpMM (native agent-scope f32 atomics, L2-resident
// accumulators) then dense [50000 x 384] @ [384 x 128] GEMM with
// V_WMMA_F32_16X16X4_F32 (fp32 matches the fp32 reference exactly).

#define N_NODES 50000
#define N_EDGES 800000
#define IN_F    96
#define OUT_F   128
#define K_CHEB  4
#define KDIM    (IN_F * K_CHEB)   // 384

typedef __attribute__((ext_vector_type(2))) float v2f;
typedef __attribute__((ext_vector_type(8))) float v8f;

// y[i] = scale * src[i]  (src==nullptr -> 0)
__global__ void cheb_init_kernel(float* __restrict__ y,
                                 const float* __restrict__ src,
                                 float scale, int n) {
    int i = blockIdx.x * blockDim.x + threadIdx.x;
    if (i < n) y[i] = src ? scale * src[i] : 0.0f;
}

// One wave32 per edge: y[row, f] += scale * vals[e] * v[col, f], f = 0..95.
// Agent-scope relaxed fetch_add lowers to native GLOBAL_ATOMIC_ADD_F32
// (no CAS retry loop); the 19.2 MB accumulator stays resident in 192 MB L2.
__global__ __launch_bounds__(256)
void cheb_spmm_scatter(const int* __restrict__ rows,
                       const int* __restrict__ cols,
                       const float* __restrict__ vals,
                       const float* __restrict__ v,
                       float* y, float scale) {
    int e    = (blockIdx.x * blockDim.x + threadIdx.x) >> 5;
    int lane = threadIdx.x & 31;
    if (e >= N_EDGES) return;
    int   r = rows[e];
    int   c = cols[e];
    float a = scale * vals[e];
    const float* src = v + (long long)c * IN_F;
    float*       dst = y + (long long)r * IN_F;
    #pragma unroll
    for (int f = lane; f < IN_F; f += 32) {
        __hip_atomic_fetch_add(&dst[f], a * src[f],
                               __ATOMIC_RELAXED, __HIP_MEMORY_SCOPE_AGENT);
    }
}

// Block = 16 rows x 128 cols of output. 8 waves, each owns one 16x16 tile.
// A tile (16 x 384, K-interleaved cheb layout) staged in LDS, shared by waves.
// 96 WMMA f32 16x16x4 ops per wave over the K=384 reduction.
__global__ __launch_bounds__(256)
void cheb_gemm_wmma(const float* __restrict__ T0,
                    const float* __restrict__ T1,
                    const float* __restrict__ T2,
                    const float* __restrict__ T3,
                    const float* __restrict__ W,     // [OUT_F, KDIM] row-major
                    const float* __restrict__ bias,  // [OUT_F]
                    float* __restrict__ out) {       // [N_NODES, OUT_F]
    __shared__ float As[16][KDIM];   // 24 KB: As[m][f*4+k] = T_k[m_base+m, f]

    const int m_base = blockIdx.x * 16;
    const int tid    = threadIdx.x;

    const float* Ts[K_CHEB] = {T0, T1, T2, T3};
    #pragma unroll
    for (int kk = 0; kk < K_CHEB; kk++) {
        const float* T = Ts[kk];
        for (int i = tid; i < 16 * IN_F; i += 256) {
            int m = i / IN_F;
            int f = i - m * IN_F;
            As[m][f * 4 + kk] = T[(long long)(m_base + m) * IN_F + f];
        }
    }
    __syncthreads();

    const int wave   = tid >> 5;        // 0..7 -> output column tile
    const int lane   = tid & 31;
    const int n_base = wave * 16;
    const int half   = lane >> 4;       // 0: K={0,1}; 1: K={2,3}
    const int lm     = lane & 15;       // A: row index; B: column index

    // A frag: lane holds row lm, 2 consecutive K at (f*4 + half*2)
    const float* arow = &As[lm][half * 2];
    // B frag: B[k][n] = W[n][k]; lane holds col n_base+lm, same K pair
    const float* wrow = W + (long long)(n_base + lm) * KDIM + half * 2;

    v8f acc = {};
    #pragma unroll 4
    for (int f = 0; f < IN_F; f++) {
        v2f a = *(const v2f*)(arow + f * 4);
        v2f b = *(const v2f*)(wrow + f * 4);
        acc = __builtin_amdgcn_wmma_f32_16x16x4_f32(
            /*neg_a=*/false, a, /*neg_b=*/false, b,
            /*c_mod=*/(short)0, acc, /*reuse_a=*/false, /*reuse_b=*/false);
    }

    // D layout: VGPR r, lane l -> row (r + 8*half), col lm
    const int   n  = n_base + lm;
    const float bv = bias[n];
    #pragma unroll
    for (int r = 0; r < 8; r++) {
        int m = m_base + r + half * 8;
        out[(long long)m * OUT_F + n] = acc[r] + bv;
    }
}

extern "C" void kernel_launch(void* const* d_in, const int* in_sizes, int n_in,
                              void* d_out, int out_size, void* d_ws, size_t ws_size,
                              hipStream_t stream) {
    const float* x    = (const float*)d_in[0];  // [N_NODES, IN_F]
    const int*   rows = (const int*)  d_in[1];  // [N_EDGES]
    const int*   cols = (const int*)  d_in[2];  // [N_EDGES]
    const float* vals = (const float*)d_in[3];  // [N_EDGES]
    const float* W    = (const float*)d_in[4];  // [OUT_F, IN_F*K]
    const float* b    = (const float*)d_in[5];  // [OUT_F]
    float*       out  = (float*)d_out;          // [N_NODES, OUT_F]

    const int nelem = N_NODES * IN_F;
    float* T1 = (float*)d_ws;
    float* T2 = T1 + nelem;
    float* T3 = T2 + nelem;

    dim3 blk(256);
    const int ginit = (nelem + 255) / 256;
    const int gscat = (N_EDGES * 32 + 255) / 256;   // one wave32 per edge

    // T1 = L x
    cheb_init_kernel<<<ginit, blk, 0, stream>>>(T1, nullptr, 0.0f, nelem);
    cheb_spmm_scatter<<<gscat, blk, 0, stream>>>(rows, cols, vals, x, T1, 1.0f);

    // T2 = 2 L T1 - x   (init accumulator to -x, scatter with scale 2)
    cheb_init_kernel<<<ginit, blk, 0, stream>>>(T2, x, -1.0f, nelem);
    cheb_spmm_scatter<<<gscat, blk, 0, stream>>>(rows, cols, vals, T1, T2, 2.0f);

    // T3 = 2 L T2 - T1
    cheb_init_kernel<<<ginit, blk, 0, stream>>>(T3, T1, -1.0f, nelem);
    cheb_spmm_scatter<<<gscat, blk, 0, stream>>>(rows, cols, vals, T2, T3, 2.0f);

    // out = cheb @ W.T + b   (50000 = 3125 * 16 rows exactly)
    cheb_gemm_wmma<<<N_NODES / 16, blk, 0, stream>>>(x, T1, T2, T3, W, b, out);
}